// Net_77214922047879
// MI455X (gfx1250) — compile-verified
//
#include <hip/hip_runtime.h>
#include <hip/hip_bf16.h>

typedef float v2f __attribute__((ext_vector_type(2)));
typedef float v8f __attribute__((ext_vector_type(8)));

#define HEADS 4
#define HID   64
#define F     256   // HEADS*HID
#define DIN   128
#define SLOPE 0.01f
#define EPS   1e-5f

// ---------------------------------------------------------------------------
// out[N,256] = X[N,K] @ W^T + b   (W is [256,K] row-major, out-feature major)
// One wave per 16x16 output tile, f32 WMMA 16x16x4, K-loop step 4.
// X strip (16 x K) staged in LDS with padded stride to avoid bank conflicts.
// ---------------------------------------------------------------------------
template <int K>
__global__ void gemm_wmma(const float* __restrict__ X,
                          const float* __restrict__ W,
                          const float* __restrict__ bias,
                          float* __restrict__ out, int N) {
    __shared__ float sx[16 * (K + 2)];
    const int tid  = threadIdx.x;
    const int lane = tid & 31;
    const int wave = tid >> 5;
    const int row0 = blockIdx.x * 16;
    const int col0 = (blockIdx.y * 8 + wave) * 16;

    // cooperative load of 16 rows of X into LDS (coalesced along K)
    for (int idx = tid; idx < 16 * K; idx += 256) {
        const int r = idx / K;
        const int c = idx - r * K;
        const int row = row0 + r;
        sx[r * (K + 2) + c] = (row < N) ? X[row * K + c] : 0.0f;
    }
    __syncthreads();

    const int m    = lane & 15;      // A row within tile / B column within tile
    const int half = lane >> 4;      // lane-half selects K pair {0,1} vs {2,3}
    const int n    = col0 + m;       // output feature (column)

    v8f acc = {0.f, 0.f, 0.f, 0.f, 0.f, 0.f, 0.f, 0.f};
    const float* wr = W + n * K;     // row of W for this output feature

    for (int k0 = 0; k0 < K; k0 += 4) {
        // A fragment: lanes 0-15 -> K=k0,k0+1 ; lanes 16-31 -> K=k0+2,k0+3
        v2f a;
        a.x = sx[m * (K + 2) + k0 + 2 * half];
        a.y = sx[m * (K + 2) + k0 + 2 * half + 1];
        // B fragment: B[k][n] = W[n][k], same K split across lane halves
        v2f b;
        b.x = wr[k0 + 2 * half];
        b.y = wr[k0 + 2 * half + 1];
        acc = __builtin_amdgcn_wmma_f32_16x16x4_f32(
            false, a, false, b, (short)0, acc, false, false);
    }

    const float bv = bias[n];
#pragma unroll
    for (int v = 0; v < 8; ++v) {
        const int orow = row0 + v + 8 * half;   // C layout: VGPR v -> M = v + 8*half
        if (orow < N) out[orow * F + n] = acc[v] + bv;
    }
}

// ---------------------------------------------------------------------------
// Per-node attention scores: s_l[n,h] = sum_d lrelu(h[n,h,d]*att_l[h,d]), same s_r.
// One wave per (node, head); 64-wide reduction over 32 lanes.
// ---------------------------------------------------------------------------
__global__ void node_scores(const float* __restrict__ h,
                            const float* __restrict__ att, // [H, 2*HID]
                            float* __restrict__ sl, float* __restrict__ sr, int N) {
    const int idx = blockIdx.x * 8 + (threadIdx.x >> 5);
    if (idx >= N * HEADS) return;
    const int lane = threadIdx.x & 31;
    const int node = idx >> 2;
    const int hd   = idx & 3;

    const float* row = h + node * F + hd * HID;
    const float* al  = att + hd * (2 * HID);
    const float* ar  = al + HID;

    float vl = 0.f, vr = 0.f;
#pragma unroll
    for (int i = lane; i < HID; i += 32) {
        const float xv = row[i];
        const float a = xv * al[i];
        const float b = xv * ar[i];
        vl += (a > 0.f) ? a : SLOPE * a;
        vr += (b > 0.f) ? b : SLOPE * b;
    }
#pragma unroll
    for (int off = 16; off > 0; off >>= 1) {
        vl += __shfl_down(vl, off, 32);
        vr += __shfl_down(vr, off, 32);
    }
    if (lane == 0) { sl[idx] = vl; sr[idx] = vr; }
}

// ---------------------------------------------------------------------------
// Edge scatter: one wave per edge.  dest = e0[e], src = e1[e].
// s_h = exp((sl[dest,h]+sr[src,h]) / (2*HID)); num[dest] += h[src]*s; den[dest] += s.
// num/den are L2-resident (51 MB / 0.8 MB << 192 MB) -> atomics stay on-chip.
// ---------------------------------------------------------------------------
__global__ void edge_scatter(const int* __restrict__ e_dst,   // e[0]
                             const int* __restrict__ e_src,   // e[1]
                             const float* __restrict__ hfeat,
                             const float* __restrict__ sl,
                             const float* __restrict__ sr,
                             float* __restrict__ num,
                             float* __restrict__ den, int E) {
    const int eid = blockIdx.x * 8 + (threadIdx.x >> 5);
    if (eid >= E) return;
    const int lane = threadIdx.x & 31;
    const int u = e_dst[eid];
    const int v = e_src[eid];

    const float inv = 1.0f / (2.0f * (float)HID);
    float sh[HEADS];
#pragma unroll
    for (int h = 0; h < HEADS; ++h)
        sh[h] = __expf((sl[u * HEADS + h] + sr[v * HEADS + h]) * inv);

    if (lane == 0) {
        atomicAdd(&den[u * HEADS + 0], sh[0]);
        atomicAdd(&den[u * HEADS + 1], sh[1]);
        atomicAdd(&den[u * HEADS + 2], sh[2]);
        atomicAdd(&den[u * HEADS + 3], sh[3]);
    }

    const float* src = hfeat + (size_t)v * F;
    float*       dst = num   + (size_t)u * F;
#pragma unroll
    for (int i = 0; i < 8; ++i) {            // head index = i>>1 (compile-time)
        const int f = lane + 32 * i;
        atomicAdd(&dst[f], src[f] * sh[i >> 1]);
    }
}

// ---------------------------------------------------------------------------
// x1 = lrelu(layernorm(num/den)) per node, block-per-node (256 threads).
// Safe in-place over num.
// ---------------------------------------------------------------------------
__global__ void norm_ln_lrelu(const float* __restrict__ num,
                              const float* __restrict__ den,
                              const float* __restrict__ gamma,
                              const float* __restrict__ beta,
                              float* __restrict__ x1, int N) {
    const int n   = blockIdx.x;
    const int tid = threadIdx.x;
    __shared__ float red[18];

    const float v = num[n * F + tid] / den[n * HEADS + (tid >> 6)];
    float s = v, q = v * v;
#pragma unroll
    for (int off = 16; off > 0; off >>= 1) {
        s += __shfl_down(s, off, 32);
        q += __shfl_down(q, off, 32);
    }
    const int wave = tid >> 5;
    if ((tid & 31) == 0) { red[wave] = s; red[8 + wave] = q; }
    __syncthreads();
    if (tid == 0) {
        float S = 0.f, Q = 0.f;
#pragma unroll
        for (int i = 0; i < 8; ++i) { S += red[i]; Q += red[8 + i]; }
        const float mu = S * (1.0f / F);
        red[16] = mu;
        red[17] = Q * (1.0f / F) - mu * mu;
    }
    __syncthreads();
    const float mu = red[16];
    const float rv = rsqrtf(red[17] + EPS);
    const float y  = (v - mu) * rv * gamma[tid] + beta[tid];
    x1[n * F + tid] = (y > 0.f) ? y : SLOPE * y;
}

// ---------------------------------------------------------------------------
// out[n,d] = mean_h num[n, h*64+d] / den[n,h]
// ---------------------------------------------------------------------------
__global__ void head_mean(const float* __restrict__ num,
                          const float* __restrict__ den,
                          float* __restrict__ out, int N) {
    const int id = blockIdx.x * blockDim.x + threadIdx.x;
    if (id >= N * HID) return;
    const int n = id >> 6;
    const int d = id & 63;
    float acc = 0.f;
#pragma unroll
    for (int h = 0; h < HEADS; ++h)
        acc += num[n * F + h * HID + d] / den[n * HEADS + h];
    out[id] = acc * (1.0f / HEADS);
}

extern "C" void kernel_launch(void* const* d_in, const int* in_sizes, int n_in,
                              void* d_out, int out_size, void* d_ws, size_t ws_size,
                              hipStream_t stream) {
    const float* x     = (const float*)d_in[0];
    const int*   e     = (const int*)d_in[1];
    const float* W0    = (const float*)d_in[2];
    const float* b0    = (const float*)d_in[3];
    const float* W1    = (const float*)d_in[4];
    const float* b1    = (const float*)d_in[5];
    const float* att0  = (const float*)d_in[6];
    const float* att1  = (const float*)d_in[7];
    const float* gamma = (const float*)d_in[8];
    const float* beta  = (const float*)d_in[9];
    float*       out   = (float*)d_out;

    const int N = in_sizes[0] / DIN;
    const int E = in_sizes[1] / 2;

    float* hbuf = (float*)d_ws;          // h0, then reused for h1   (N*F)
    float* nbuf = hbuf + (size_t)N * F;  // num0 -> x1 in place      (N*F)
    float* mbuf = nbuf + (size_t)N * F;  // num1                     (N*F)
    float* den0 = mbuf + (size_t)N * F;  // (N*H)
    float* den1 = den0 + (size_t)N * HEADS;
    float* sl   = den1 + (size_t)N * HEADS;
    float* sr   = sl   + (size_t)N * HEADS;

    // zero all accumulators (nbuf, mbuf, den0, den1 are contiguous)
    hipMemsetAsync(nbuf, 0,
                   ((size_t)2 * N * F + (size_t)2 * N * HEADS) * sizeof(float),
                   stream);

    const int rowTiles = (N + 15) / 16;
    const int nhWaves  = (N * HEADS + 7) / 8;
    const int eWaves   = (E + 7) / 8;

    // ---- layer 0 ----
    gemm_wmma<DIN><<<dim3(rowTiles, 2), 256, 0, stream>>>(x, W0, b0, hbuf, N);
    node_scores<<<nhWaves, 256, 0, stream>>>(hbuf, att0, sl, sr, N);
    edge_scatter<<<eWaves, 256, 0, stream>>>(e, e + E, hbuf, sl, sr, nbuf, den0, E);
    norm_ln_lrelu<<<N, 256, 0, stream>>>(nbuf, den0, gamma, beta, nbuf, N);

    // ---- layer 1 ----
    gemm_wmma<F><<<dim3(rowTiles, 2), 256, 0, stream>>>(nbuf, W1, b1, hbuf, N);
    node_scores<<<nhWaves, 256, 0, stream>>>(hbuf, att1, sl, sr, N);
    edge_scatter<<<eWaves, 256, 0, stream>>>(e, e + E, hbuf, sl, sr, mbuf, den1, E);

    head_mean<<<(N * HID + 255) / 256, 256, 0, stream>>>(mbuf, den1, out, N);
}